// g_49323404427297
// MI455X (gfx1250) — compile-verified
//
#include <hip/hip_runtime.h>

// Bilinear backward warp, MI455X (gfx1250, wave32).
// Memory-bound gather kernel: ~134 MB compulsory traffic -> ~5.8 us at 23.3 TB/s.
// WMMA is structurally inapplicable (per-output gather addresses + per-output
// weights have no shared-operand matmul structure). Levers: NT temporal hints
// for streamed flow/out (preserve 192MB L2 for img gather reuse), explicit
// inline-asm global_load_b64/b96 clustering with hand-placed s_wait_loadcnt so
// both flow loads and all 8 corner gathers are in flight concurrently.
// (sched_barrier proved insufficient: ISel may still sink non-volatile loads.)

typedef float f32x2 __attribute__((ext_vector_type(2)));
typedef float f32x3 __attribute__((ext_vector_type(3)));

// NT b64 load (flow is streamed exactly once; keep it out of the gather cache).
__device__ inline f32x2 load_b64_nt(const float* p) {
    f32x2 r;
    asm volatile("global_load_b64 %0, %1, off th:TH_LOAD_NT"
                 : "=v"(r) : "v"(p) : "memory");
    return r;
}

// Default (RT) b96 load: img gathers want caching (line reuse ~40x).
__device__ inline f32x3 load_b96(const float* p) {
    f32x3 r;
    asm volatile("global_load_b96 %0, %1, off"
                 : "=v"(r) : "v"(p) : "memory");
    return r;
}

__device__ inline int iclamp(int v, int lo, int hi) {
    v = v < lo ? lo : v;
    return v > hi ? hi : v;
}

// Per-pixel address computation + bilinear weights (after flow is loaded).
template <int W, int H, int C>
__device__ inline void addr_weights(unsigned p, float dx, float dy,
                                    const float* __restrict__ img,
                                    const float*& pa, const float*& pb,
                                    const float*& pc, const float*& pd,
                                    float& wa, float& wb, float& wc, float& wd) {
    int xi = (int)(p & (unsigned)(W - 1));
    int yi = (int)((p >> 9) & (unsigned)(H - 1));   // W == 512
    int bi = (int)(p >> 18);                        // H*W == 2^18

    float x = (float)xi + dx;
    float y = (float)yi + dy;

    float x0f = floorf(x);
    float y0f = floorf(y);
    float x1f = x0f + 1.0f;
    float y1f = y0f + 1.0f;

    // Same weight expressions as the reference (unclipped corner coords).
    wa = (x1f - x) * (y1f - y);
    wb = (x1f - x) * (y - y0f);
    wc = (x - x0f) * (y1f - y);
    wd = (x - x0f) * (y - y0f);

    int x0i = iclamp((int)x0f, 0, W - 1);
    int x1i = iclamp((int)x1f, 0, W - 1);
    int y0i = iclamp((int)y0f, 0, H - 1);
    int y1i = iclamp((int)y1f, 0, H - 1);

    const float* base = img + (size_t)bi * (size_t)(H * W * C);
    pa = base + (y0i * W + x0i) * C;   // top-left
    pb = base + (y1i * W + x0i) * C;   // bottom-left
    pc = base + (y0i * W + x1i) * C;   // top-right
    pd = base + (y1i * W + x1i) * C;   // bottom-right
}

__global__ __launch_bounds__(256) void warp_bilinear_kernel(
    const float* __restrict__ img,
    const float* __restrict__ flow,
    float* __restrict__ out,
    unsigned npix) {
    constexpr int W = 512, H = 512, C = 3;

    // 2 pixels per thread, interleaved by 256: one block covers a contiguous
    // 512-pixel row (uniform b,y across the block).
    unsigned p0 = blockIdx.x * 512u + threadIdx.x;
    unsigned p1 = p0 + 256u;
    bool ok0 = p0 < npix;
    bool ok1 = p1 < npix;
    // Clamp indices for loads instead of branching: keeps ONE basic block.
    unsigned q0 = ok0 ? p0 : 0u;
    unsigned q1 = ok1 ? p1 : 0u;

    // ---- Phase 1: both flow loads in flight, then ONE wait ----
    f32x2 f0 = load_b64_nt(flow + 2u * q0);
    f32x2 f1 = load_b64_nt(flow + 2u * q1);
    // Tie loaded values through the wait: the compiler cannot let any consumer
    // of f0/f1 execute before this s_wait_loadcnt (no HW scoreboard on loads).
    asm volatile("s_wait_loadcnt 0x0" : "+v"(f0), "+v"(f1)::"memory");

    // ---- Phase 2: address + weight math, then all 8 gathers, ONE wait ----
    const float *pa0, *pb0, *pc0, *pd0, *pa1, *pb1, *pc1, *pd1;
    float wa0, wb0, wc0, wd0, wa1, wb1, wc1, wd1;
    addr_weights<W, H, C>(q0, f0.x, f0.y, img, pa0, pb0, pc0, pd0, wa0, wb0, wc0, wd0);
    addr_weights<W, H, C>(q1, f1.x, f1.y, img, pa1, pb1, pc1, pd1, wa1, wb1, wc1, wd1);

    f32x3 A0 = load_b96(pa0);
    f32x3 B0 = load_b96(pb0);
    f32x3 C0 = load_b96(pc0);
    f32x3 D0 = load_b96(pd0);
    f32x3 A1 = load_b96(pa1);
    f32x3 B1 = load_b96(pb1);
    f32x3 C1 = load_b96(pc1);
    f32x3 D1 = load_b96(pd1);
    asm volatile("s_wait_loadcnt 0x0"
                 : "+v"(A0), "+v"(B0), "+v"(C0), "+v"(D0),
                   "+v"(A1), "+v"(B1), "+v"(C1), "+v"(D1)
                 ::"memory");

    // ---- Phase 3: weighted sums + predicated NT stores ----
    float r00 = wa0 * A0.x + wb0 * B0.x + wc0 * C0.x + wd0 * D0.x;
    float r01 = wa0 * A0.y + wb0 * B0.y + wc0 * C0.y + wd0 * D0.y;
    float r02 = wa0 * A0.z + wb0 * B0.z + wc0 * C0.z + wd0 * D0.z;
    float r10 = wa1 * A1.x + wb1 * B1.x + wc1 * C1.x + wd1 * D1.x;
    float r11 = wa1 * A1.y + wb1 * B1.y + wc1 * C1.y + wd1 * D1.y;
    float r12 = wa1 * A1.z + wb1 * B1.z + wc1 * C1.z + wd1 * D1.z;

    if (ok0) {
        float* po = out + (size_t)p0 * (size_t)C;
        __builtin_nontemporal_store(r00, po);
        __builtin_nontemporal_store(r01, po + 1);
        __builtin_nontemporal_store(r02, po + 2);
    }
    if (ok1) {
        float* po = out + (size_t)p1 * (size_t)C;
        __builtin_nontemporal_store(r10, po);
        __builtin_nontemporal_store(r11, po + 1);
        __builtin_nontemporal_store(r12, po + 2);
    }
}

extern "C" void kernel_launch(void* const* d_in, const int* in_sizes, int n_in,
                              void* d_out, int out_size, void* d_ws, size_t ws_size,
                              hipStream_t stream) {
    const float* img  = (const float*)d_in[0];   // [16,512,512,3] f32
    const float* flow = (const float*)d_in[1];   // [16,512,512,2] f32
    float* out = (float*)d_out;                  // [16,512,512,3] f32

    unsigned npix = (unsigned)(in_sizes[1] / 2); // B*H*W
    unsigned blocks = (npix + 511u) / 512u;      // 2 px/thread, 256 threads/block
    warp_bilinear_kernel<<<blocks, 256, 0, stream>>>(img, flow, out, npix);
}